// SlimeMoldEncoder_14070312862440
// MI455X (gfx1250) — compile-verified
//
#include <hip/hip_runtime.h>
#include <hip/hip_bf16.h>
#include <math.h>

typedef __attribute__((ext_vector_type(2))) float v2f;
typedef __attribute__((ext_vector_type(8))) float v8f;

// ---------------------------------------------------------------------------
// V_WMMA_F32_16X16X4_F32 fragment layout (wave32):
//   A (16x4, MxK):  lane L -> row M = L&15;  v.x = A[M][k + 2*(L>=16)],
//                                            v.y = A[M][k + 2*(L>=16) + 1]
//   B (4x16, KxN):  lane L -> col N = L&15;  v.x = B[k + 2*(L>=16)    ][N],
//                                            v.y = B[k + 2*(L>=16) + 1][N]
//   C/D (16x16):    lane L -> col N = L&15;  vgpr r -> row M = r + 8*(L>=16)
//
// Register-tiled strip: one wave accumulates MT 16-row tiles x 4 16-col tiles
// of A[.. x 1024] @ W[1024 x 1024]. Per K=4 step: MT A-b64 + 8 B-b32 loads
// feed 4*MT WMMAs (A and B fragments each reused 4x / MT x).
// ---------------------------------------------------------------------------
template <int MT>
__device__ __forceinline__ void wmma_block_f32(
    const float* __restrict__ A,      // [.. x 1024] row-major
    const float* __restrict__ W,      // [1024 x 1024] row-major
    int m0, int n0, int l15, int koff,
    v8f acc[MT][4])
{
    const float* Arow[MT];
#pragma unroll
    for (int mt = 0; mt < MT; ++mt)
        Arow[mt] = A + (size_t)(m0 + mt * 16 + l15) * 1024;

#pragma unroll 2
    for (int k = 0; k < 1024; k += 4) {
        v2f a[MT];
#pragma unroll
        for (int mt = 0; mt < MT; ++mt) {
            a[mt].x = Arow[mt][k + koff];
            a[mt].y = Arow[mt][k + koff + 1];
        }
        const float* w0 = W + (size_t)(k + koff) * 1024 + n0 + l15;
        v2f b[4];
#pragma unroll
        for (int nt = 0; nt < 4; ++nt) {
            b[nt].x = w0[nt * 16];
            b[nt].y = w0[nt * 16 + 1024];
        }
#pragma unroll
        for (int mt = 0; mt < MT; ++mt)
#pragma unroll
            for (int nt = 0; nt < 4; ++nt)
                acc[mt][nt] = __builtin_amdgcn_wmma_f32_16x16x4_f32(
                    false, a[mt], false, b[nt], (short)0, acc[mt][nt], false, false);
    }
}

// ---------------------------------------------------------------------------
// C[M x 1024] = A[M x 1024] @ W[1024 x 1024] + bias[1024]
// Block = 128 threads (4 waves stacked along M, each wave owns 64x64).
// Grid = (1024/64, M/256). Used for the batched input projection
// (src@W_in + b_h) and the batched output projection (H@W_out + b_o).
// ---------------------------------------------------------------------------
__global__ __launch_bounds__(128) void gemm_bias_kernel(
    const float* __restrict__ A,
    const float* __restrict__ W,
    const float* __restrict__ bias,
    float* __restrict__ C,
    int M)
{
    const int lane = threadIdx.x & 31;
    const int wave = threadIdx.x >> 5;
    const int l15  = lane & 15;
    const int half = lane >> 4;
    const int koff = half * 2;

    const int n0 = blockIdx.x * 64;
    const int m0 = (blockIdx.y * 4 + wave) * 64;
    if (m0 >= M) return;   // wave-uniform; EXEC stays all-ones inside waves

    v8f acc[4][4] = {};
    wmma_block_f32<4>(A, W, m0, n0, l15, koff, acc);

    float bj[4];
#pragma unroll
    for (int nt = 0; nt < 4; ++nt) bj[nt] = bias[n0 + nt * 16 + l15];

#pragma unroll
    for (int mt = 0; mt < 4; ++mt) {
        float* Cbase = C + (size_t)(m0 + mt * 16 + half * 8) * 1024 + n0 + l15;
#pragma unroll
        for (int r = 0; r < 8; ++r) {
            float* Crow = Cbase + (size_t)r * 1024;
#pragma unroll
            for (int nt = 0; nt < 4; ++nt)
                Crow[nt * 16] = acc[mt][nt][r] + bj[nt];
        }
    }
}

// ---------------------------------------------------------------------------
// One recurrent step, in place:
//   buf_t[32 x 1024] <- tanh(buf_t + h_prev @ W_rec)
// buf_t already holds x_t@W_in + b_h. 16 waves (grid=4, block=128):
// each wave owns a 32(M) x 64(N) block of the state (2x4 accumulators).
// ---------------------------------------------------------------------------
__global__ __launch_bounds__(128) void rnn_step_kernel(
    float* __restrict__ buf_t,
    const float* __restrict__ h_prev,
    const float* __restrict__ Wrec,
    int first)
{
    const int lane = threadIdx.x & 31;
    const int wave_id = blockIdx.x * 4 + (threadIdx.x >> 5);
    const int l15  = lane & 15;
    const int half = lane >> 4;
    const int koff = half * 2;

    const int n0 = wave_id * 64;

    v8f acc[2][4] = {};
    if (!first)
        wmma_block_f32<2>(h_prev, Wrec, 0, n0, l15, koff, acc);

#pragma unroll
    for (int mt = 0; mt < 2; ++mt) {
        float* base = buf_t + (size_t)(mt * 16 + half * 8) * 1024 + n0 + l15;
#pragma unroll
        for (int r = 0; r < 8; ++r) {
            float* row = base + (size_t)r * 1024;
#pragma unroll
            for (int nt = 0; nt < 4; ++nt)
                row[nt * 16] = tanhf(row[nt * 16] + acc[mt][nt][r]);
        }
    }
}

extern "C" void kernel_launch(void* const* d_in, const int* in_sizes, int n_in,
                              void* d_out, int out_size, void* d_ws, size_t ws_size,
                              hipStream_t stream) {
    const float* src   = (const float*)d_in[0];  // [S, B, D] = [1024, 32, 1024]
    const float* W_in  = (const float*)d_in[1];  // [D, D]
    const float* W_rec = (const float*)d_in[2];  // [D, D]
    const float* W_out = (const float*)d_in[3];  // [D, D]
    const float* b_h   = (const float*)d_in[4];  // [D]
    const float* b_o   = (const float*)d_in[5];  // [D]
    float* out = (float*)d_out;                  // [S, B, D]
    float* buf = (float*)d_ws;                   // S*B*D floats = 128 MB scratch

    const int S = 1024, Bb = 32, Dm = 1024;
    const int M = S * Bb;                        // 32768 flattened rows

    // Phase 1: buf = src @ W_in + b_h   (fully parallel over all timesteps)
    dim3 gg(Dm / 64, M / 256);                   // (16, 128)
    gemm_bias_kernel<<<gg, 128, 0, stream>>>(src, W_in, b_h, buf, M);

    // Phase 2: sequential recurrence; buf[t] becomes h_t in place.
    for (int t = 0; t < S; ++t) {
        rnn_step_kernel<<<4, 128, 0, stream>>>(
            buf + (size_t)t * Bb * Dm,
            t ? (buf + (size_t)(t - 1) * Bb * Dm) : buf,
            W_rec, (t == 0) ? 1 : 0);
    }

    // Phase 3: out = H @ W_out + b_o    (fully parallel over all timesteps)
    gemm_bias_kernel<<<gg, 128, 0, stream>>>(buf, W_out, b_o, out, M);
}